// GraphEncoder_7275674600336
// MI455X (gfx1250) — compile-verified
//
#include <hip/hip_runtime.h>
#include <hip/hip_bf16.h>
#include <math.h>

// ---------------------------------------------------------------------------
// GCN graph encoder for MI455X (gfx1250, wave32).
//   h = x @ W1            -> V_WMMA_F32_16X16X4_F32 (full fp32 matrix pipe)
//   agg = A_norm @ h      -> edge scatter with global f32 atomics (L2 resident)
//   pool + head           -> LDS-reduced per-graph sums, tiny fp32 head
// ---------------------------------------------------------------------------

typedef float v2f __attribute__((ext_vector_type(2)));
typedef float v8f __attribute__((ext_vector_type(8)));

#define F1 64      // hidden width (W1 cols)
#define F0 128     // input width
#define F2 32      // output width (W2 cols)
#define NG 64      // NUM_GRAPHS
#define SLOPE 0.01f

// --- 1) init workspace: deg=1.0 (self-loop), agg/sums/cnt = 0 ---------------
__global__ void init_kernel(float* __restrict__ agg, float* __restrict__ deg,
                            float* __restrict__ sums, float* __restrict__ cnt,
                            int N) {
  int i = blockIdx.x * blockDim.x + threadIdx.x;
  long long total = (long long)N * F1;
  if (i < total) agg[i] = 0.0f;
  if (i < N) deg[i] = 1.0f;
  if (i < NG * F1) sums[i] = 0.0f;
  if (i < NG) cnt[i] = 0.0f;
}

// --- 2) degree accumulation over edge destinations --------------------------
__global__ void deg_kernel(const long long* __restrict__ dst,
                           float* __restrict__ deg, long long E) {
  long long e = (long long)blockIdx.x * blockDim.x + threadIdx.x;
  if (e < E) atomicAdd(&deg[(int)dst[e]], 1.0f);
}

// --- 3) dinv = rsqrt(deg), in place (deg >= 1 always) -----------------------
__global__ void dinv_kernel(float* __restrict__ deg, int N) {
  int i = blockIdx.x * blockDim.x + threadIdx.x;
  if (i < N) deg[i] = rsqrtf(deg[i]);
}

// --- 4) h = x @ W1 via V_WMMA_F32_16X16X4_F32 -------------------------------
// One wave computes one 16x16 tile of h. blockDim = 128 (4 waves), the 4
// waves cover the 4 column tiles (F1=64 -> 4 tiles of 16). blockIdx.x = row
// tile. ISA f32 layouts (CDNA5 ISA 7.12.2):
//   A (16x4):  lane = M (l&15), half (l>>4) selects K+2;  VGPR0=K, VGPR1=K+1
//   B (4x16):  lane = N (l&15), half selects K+2;          VGPR0=K, VGPR1=K+1
//   C/D:       VGPR r: lanes 0-15 -> M=r, lanes 16-31 -> M=r+8, N = l&15
__global__ void gemm_h_kernel(const float* __restrict__ x,
                              const float* __restrict__ W1,
                              float* __restrict__ h, int N) {
  const int lane = threadIdx.x & 31;
  const int wave = threadIdx.x >> 5;
  const int half = lane >> 4;   // 0 or 1
  const int l15  = lane & 15;
  const int m0 = blockIdx.x * 16;
  const int n0 = wave * 16;

  int arow = m0 + l15;
  if (arow > N - 1) arow = N - 1;          // clamp loads; EXEC must stay all-1
  const float* xrow = x + (size_t)arow * F0;

  v8f c = {0.f, 0.f, 0.f, 0.f, 0.f, 0.f, 0.f, 0.f};
  for (int k0 = 0; k0 < F0; k0 += 4) {
    v2f a, b;
    a.x = xrow[k0 + 2 * half + 0];
    a.y = xrow[k0 + 2 * half + 1];
    b.x = W1[(k0 + 2 * half + 0) * F1 + n0 + l15];
    b.y = W1[(k0 + 2 * half + 1) * F1 + n0 + l15];
    c = __builtin_amdgcn_wmma_f32_16x16x4_f32(false, a, false, b,
                                              (short)0, c, false, false);
  }

#pragma unroll
  for (int r = 0; r < 8; ++r) {
    int row = m0 + r + 8 * half;
    if (row < N) h[(size_t)row * F1 + n0 + l15] = c[r];
  }
}

// --- 5) agg[dst] += dinv[src]*dinv[dst] * h[src]  (one wave per edge) -------
__global__ void edge_agg_kernel(const long long* __restrict__ src,
                                const long long* __restrict__ dst,
                                const float* __restrict__ dinv,
                                const float* __restrict__ h,
                                float* __restrict__ agg, long long E) {
  long long e = (long long)blockIdx.x * (blockDim.x >> 5) + (threadIdx.x >> 5);
  if (e >= E) return;
  const int lane = threadIdx.x & 31;
  const int s = (int)src[e];
  const int d = (int)dst[e];
  const float norm = dinv[s] * dinv[d];
  const float* hs = h + (size_t)s * F1;
  float* ad = agg + (size_t)d * F1;
  atomicAdd(&ad[lane],      norm * hs[lane]);
  atomicAdd(&ad[lane + 32], norm * hs[lane + 32]);
}

// --- 6) self-loop term + bias + LeakyReLU + per-graph pooled sums -----------
// 8 nodes per block (one wave each). batch is sorted, so blocks are almost
// always single-graph: reduce the 8 nodes in LDS, then one global atomic set.
__global__ void pool_kernel(const float* __restrict__ h,
                            const float* __restrict__ agg,
                            const float* __restrict__ dinv,
                            const float* __restrict__ b1,
                            const long long* __restrict__ batch,
                            float* __restrict__ sums, float* __restrict__ cnt,
                            int N) {
  __shared__ float lsum[F1];
  const int wave = threadIdx.x >> 5;
  const int lane = threadIdx.x & 31;
  const int node0 = blockIdx.x * 8;
  const int node = node0 + wave;
  int last = node0 + 7; if (last > N - 1) last = N - 1;
  const bool full    = (node0 + 7 < N);
  const bool uniform = full && ((int)batch[node0] == (int)batch[last]);

  if (uniform && threadIdx.x < F1) lsum[threadIdx.x] = 0.0f;
  __syncthreads();

  int g = 0;
  if (node < N) {
    g = (int)batch[node];
    const float di = dinv[node];
    const float selfn = di * di;
    const float* hn = h + (size_t)node * F1;
    const float* an = agg + (size_t)node * F1;
#pragma unroll
    for (int t = 0; t < 2; ++t) {
      const int f = lane + 32 * t;
      float v = an[f] + selfn * hn[f] + b1[f];
      float act = v > 0.0f ? v : SLOPE * v;
      if (uniform) atomicAdd(&lsum[f], act);
      else         atomicAdd(&sums[(size_t)g * F1 + f], act);
    }
    if (!uniform && lane == 0) atomicAdd(&cnt[g], 1.0f);
  }
  __syncthreads();

  if (uniform && wave == 0) {
    const int gg = (int)batch[node0];
    atomicAdd(&sums[(size_t)gg * F1 + lane],      lsum[lane]);
    atomicAdd(&sums[(size_t)gg * F1 + lane + 32], lsum[lane + 32]);
    if (lane == 0) atomicAdd(&cnt[gg], 8.0f);
  }
}

// --- 7) out = tanh((sums/cnt) @ W2 + b2), [64,64]@[64,32] -> [64,32] --------
__global__ void head_kernel(const float* __restrict__ sums,
                            const float* __restrict__ cnt,
                            const float* __restrict__ W2,
                            const float* __restrict__ b2,
                            float* __restrict__ out) {
  int idx = blockIdx.x * blockDim.x + threadIdx.x;   // 0 .. NG*F2-1
  if (idx >= NG * F2) return;
  const int g = idx / F2;
  const int j = idx % F2;
  float c = cnt[g];
  if (c < 1.0f) c = 1.0f;
  const float inv = 1.0f / c;
  float acc = b2[j];
  const float* sg = sums + (size_t)g * F1;
#pragma unroll 4
  for (int k = 0; k < F1; ++k) acc += (sg[k] * inv) * W2[k * F2 + j];
  out[(size_t)g * F2 + j] = tanhf(acc);
}

// ---------------------------------------------------------------------------
extern "C" void kernel_launch(void* const* d_in, const int* in_sizes, int n_in,
                              void* d_out, int out_size, void* d_ws, size_t ws_size,
                              hipStream_t stream) {
  const float*     x     = (const float*)d_in[0];
  const float*     W1    = (const float*)d_in[1];
  const float*     b1    = (const float*)d_in[2];
  const float*     W2    = (const float*)d_in[3];
  const float*     b2    = (const float*)d_in[4];
  const long long* eidx  = (const long long*)d_in[5];   // int64 [2, E] flat
  const long long* batch = (const long long*)d_in[6];   // int64 [N]

  const int       N = in_sizes[0] / F0;
  const long long E = (long long)in_sizes[5] / 2;
  const long long* src = eidx;        // edge_index[0]
  const long long* dst = eidx + E;    // edge_index[1]

  // workspace layout (floats): h[N*64] | agg[N*64] | dinv[N] | sums[64*64] | cnt[64]
  float* ws   = (float*)d_ws;
  float* h    = ws;
  float* agg  = h + (size_t)N * F1;
  float* deg  = agg + (size_t)N * F1;   // deg, overwritten by dinv
  float* sums = deg + N;
  float* cnt  = sums + NG * F1;

  const int initElems = N * F1;
  init_kernel<<<(initElems + 255) / 256, 256, 0, stream>>>(agg, deg, sums, cnt, N);
  deg_kernel<<<(unsigned)((E + 255) / 256), 256, 0, stream>>>(dst, deg, E);
  dinv_kernel<<<(N + 255) / 256, 256, 0, stream>>>(deg, N);
  gemm_h_kernel<<<(N + 15) / 16, 128, 0, stream>>>(x, W1, h, N);
  edge_agg_kernel<<<(unsigned)((E + 7) / 8), 256, 0, stream>>>(src, dst, deg, h, agg, E);
  pool_kernel<<<(N + 7) / 8, 256, 0, stream>>>(h, agg, deg, b1, batch, sums, cnt, N);
  head_kernel<<<8, 256, 0, stream>>>(sums, cnt, W2, b2, (float*)d_out);
}